// LSTM_65429531787831
// MI455X (gfx1250) — compile-verified
//
#include <hip/hip_runtime.h>
#include <stdint.h>

// Problem dims
#define B_    512
#define T_    1024
#define TC    1022      // T - 2 (valid conv)
#define F_    32
#define CO    64        // CNN_OUT
#define HID_  32
#define G_    128       // 4 * HID

typedef __attribute__((ext_vector_type(2))) float v2f;
typedef __attribute__((ext_vector_type(8))) float v8f;

__device__ __forceinline__ v8f wmma_f32(v2f a, v2f b, v8f c) {
  // D = A(16x4,f32) x B(4x16,f32) + C(16x16,f32)
  return __builtin_amdgcn_wmma_f32_16x16x4_f32(
      /*neg_a=*/false, a, /*neg_b=*/false, b,
      /*c_mod=*/(short)0, c, /*reuse_a=*/false, /*reuse_b=*/false);
}

__device__ __forceinline__ float sigm(float x) {
  return 1.0f / (1.0f + __expf(-x));
}

// --- CDNA5 async global->LDS copy (ASYNCcnt-tracked), 16B per lane ---------
__device__ __forceinline__ void async_b128(uint32_t lds_off, const void* gptr) {
  // VDST = LDS byte offset VGPR, VADDR = 64-bit global address (GV mode)
  asm volatile("global_load_async_to_lds_b128 %0, %1, off"
               :: "v"(lds_off), "v"((unsigned long long)(uintptr_t)gptr)
               : "memory");
}
__device__ __forceinline__ void wait_async0() {
#if __has_builtin(__builtin_amdgcn_s_wait_asynccnt)
  __builtin_amdgcn_s_wait_asynccnt(0);
#else
  asm volatile("s_wait_asynccnt 0x0" ::: "memory");
#endif
}
__device__ __forceinline__ void wait_async4() {
#if __has_builtin(__builtin_amdgcn_s_wait_asynccnt)
  __builtin_amdgcn_s_wait_asynccnt(4);
#else
  asm volatile("s_wait_asynccnt 0x4" ::: "memory");
#endif
}

// Copy one 8 KB (16 x 128 f32) timestep tile: 128 threads x 4 x 16B each.
// Generic pointer to __shared__: low 32 bits == LDS byte offset.
__device__ __forceinline__ void issue_tile(const float* g, const float* lds_buf,
                                           int tid) {
  const uint32_t l0 = (uint32_t)(uintptr_t)lds_buf + (uint32_t)tid * 16u;
  const char* gp = (const char*)g + (size_t)tid * 16;
  #pragma unroll
  for (int k = 0; k < 4; ++k)
    async_b128(l0 + (uint32_t)k * 2048u, gp + (size_t)k * 2048);
}

// ---------------------------------------------------------------------------
// K1: fused Conv1D(k=3, valid) + bias + ReLU + GEMM with W1 (+b1) -> zx1
// zx1 layout: (TC, B, 128) so the LSTM reads contiguous tiles per timestep.
// Block: 128 threads (4 waves), handles one batch row x 16 timesteps.
// ---------------------------------------------------------------------------
__global__ __launch_bounds__(128) void k_conv_zx1(
    const float* __restrict__ x, const float* __restrict__ cw,
    const float* __restrict__ cb, const float* __restrict__ W1,
    const float* __restrict__ b1, float* __restrict__ zx1)
{
  __shared__ float s_patch[18 * F_];   // input patch: 18 time rows x 32 feats
  __shared__ float s_cw[96 * CO];      // conv weights as 96x64 GEMM B
  __shared__ float s_w1[CO * G_];      // W1 64x128
  __shared__ float s_conv[16 * CO];    // relu(conv) tile 16x64

  const int b    = blockIdx.y;
  const int t0   = blockIdx.x * 16;
  const int tid  = threadIdx.x;
  const int lane = tid & 31;
  const int wave = tid >> 5;
  const int m    = lane & 15;           // A-matrix row (time row in tile)
  const int koff = (lane >> 4) << 1;    // K phase: lanes 16-31 hold K+2
  const int mb   = (lane >> 4) << 3;    // C-matrix row base (0 or 8)

  // Cooperative loads into LDS
  const float* xb = x + (size_t)b * (T_ * F_);
  for (int i = tid; i < 18 * F_; i += 128) {
    int t = t0 + (i >> 5);
    s_patch[i] = (t < T_) ? xb[(size_t)t * F_ + (i & 31)] : 0.0f;
  }
  for (int i = tid; i < 96 * CO; i += 128) s_cw[i] = cw[i];
  for (int i = tid; i < CO * G_; i += 128) s_w1[i] = W1[i];
  __syncthreads();

  // --- Conv as GEMM: A(16x96 patch) @ B(96x16 weights) per wave ---
  {
    const int n = (wave << 4) + (lane & 15);   // conv output channel
    v8f acc;
    const float bias = cb[n];
    #pragma unroll
    for (int i = 0; i < 8; ++i) acc[i] = bias;
    #pragma unroll
    for (int ks = 0; ks < 24; ++ks) {
      const int kk = (ks << 2) + koff;          // even -> v2f stays in-row
      v2f a = *(const v2f*)&s_patch[(m + (kk >> 5)) * F_ + (kk & 31)];
      v2f w; w.x = s_cw[kk * CO + n]; w.y = s_cw[(kk + 1) * CO + n];
      acc = wmma_f32(a, w, acc);
    }
    #pragma unroll
    for (int i = 0; i < 8; ++i)
      s_conv[(mb + i) * CO + n] = fmaxf(acc[i], 0.0f);   // ReLU
  }
  __syncthreads();

  // --- zx1 = relu(conv) @ W1 + b1 : 16x64 @ 64x128, 32 cols per wave ---
  #pragma unroll
  for (int nt = 0; nt < 2; ++nt) {
    const int gn = (wave << 5) + (nt << 4) + (lane & 15);
    v8f acc;
    const float bias = b1[gn];
    #pragma unroll
    for (int i = 0; i < 8; ++i) acc[i] = bias;
    #pragma unroll
    for (int ks = 0; ks < 16; ++ks) {
      const int kk = (ks << 2) + koff;
      v2f a = *(const v2f*)&s_conv[m * CO + kk];
      v2f w; w.x = s_w1[kk * G_ + gn]; w.y = s_w1[(kk + 1) * G_ + gn];
      acc = wmma_f32(a, w, acc);
    }
    #pragma unroll
    for (int i = 0; i < 8; ++i) {
      const int t = t0 + mb + i;
      if (t < TC) zx1[((size_t)t * B_ + b) * G_ + gn] = acc[i];
    }
  }
}

// ---------------------------------------------------------------------------
// K2: LSTM1 over full sequence; fuses projection zx2 = h @ W2 + b2.
// Block: 128 threads, 16 batch rows; h/c resident in LDS; zx tiles streamed
// through a double-buffered async global->LDS pipeline (ASYNCcnt).
// ---------------------------------------------------------------------------
__global__ __launch_bounds__(128) void k_lstm1(
    const float* __restrict__ zx1, const float* __restrict__ U1,
    const float* __restrict__ W2, const float* __restrict__ b2,
    float* __restrict__ zx2)
{
  __shared__ float s_zx[2][16 * G_];   // double-buffered zx tiles (2 x 8 KB)
  __shared__ float s_h[16 * HID_];
  __shared__ float s_c[16 * HID_];
  __shared__ float s_z[16 * G_];
  __shared__ float s_u[HID_ * G_];
  __shared__ float s_w2[HID_ * G_];

  const int b0   = blockIdx.x * 16;
  const int tid  = threadIdx.x;
  const int lane = tid & 31;
  const int wave = tid >> 5;
  const int m    = lane & 15;
  const int koff = (lane >> 4) << 1;
  const int mb   = (lane >> 4) << 3;

  for (int i = tid; i < HID_ * G_; i += 128) { s_u[i] = U1[i]; s_w2[i] = W2[i]; }
  for (int i = tid; i < 16 * HID_; i += 128) { s_h[i] = 0.0f; s_c[i] = 0.0f; }

  const float b2v0 = b2[(wave << 5) + (lane & 15)];
  const float b2v1 = b2[(wave << 5) + 16 + (lane & 15)];

  issue_tile(zx1 + (size_t)b0 * G_, s_zx[0], tid);   // tile t=0
  __syncthreads();

  for (int t = 0; t < TC; ++t) {
    const float* zl = s_zx[t & 1];
    if (t + 1 < TC) {
      issue_tile(zx1 + ((size_t)(t + 1) * B_ + b0) * G_, s_zx[(t + 1) & 1], tid);
      wait_async4();   // in-order: retires current tile, next stays in flight
    } else {
      wait_async0();
    }
    __syncthreads();   // (a) all waves' async chunks visible

    // z = zx1[t] + h @ U1   (16x32 @ 32x128, 32 cols per wave)
    #pragma unroll
    for (int nt = 0; nt < 2; ++nt) {
      const int gn = (wave << 5) + (nt << 4) + (lane & 15);
      v8f acc;
      #pragma unroll
      for (int i = 0; i < 8; ++i) acc[i] = zl[(mb + i) * G_ + gn];
      #pragma unroll
      for (int ks = 0; ks < 8; ++ks) {
        const int kk = (ks << 2) + koff;
        v2f a = *(const v2f*)&s_h[m * HID_ + kk];
        v2f w; w.x = s_u[kk * G_ + gn]; w.y = s_u[(kk + 1) * G_ + gn];
        acc = wmma_f32(a, w, acc);
      }
      #pragma unroll
      for (int i = 0; i < 8; ++i) s_z[(mb + i) * G_ + gn] = acc[i];
    }
    __syncthreads();   // (b)

    // gates + state update (512 (m,j) pairs, 4 per thread)
    #pragma unroll
    for (int k = 0; k < 4; ++k) {
      const int idx = tid + (k << 7);         // == mm*32 + j
      const int mm = idx >> 5, j = idx & 31;
      const float zi = s_z[mm * G_ + j];
      const float zf = s_z[mm * G_ + HID_ + j];
      const float zg = s_z[mm * G_ + 2 * HID_ + j];
      const float zo = s_z[mm * G_ + 3 * HID_ + j];
      const float c = sigm(zf) * s_c[idx] + sigm(zi) * tanhf(zg);
      s_c[idx] = c;
      s_h[idx] = sigm(zo) * tanhf(c);
    }
    __syncthreads();   // (c)

    // zx2[t] = h @ W2 + b2  (feeds LSTM2; avoids materializing h1)
    float* z2 = zx2 + ((size_t)t * B_ + b0) * G_;
    #pragma unroll
    for (int nt = 0; nt < 2; ++nt) {
      const int gn = (wave << 5) + (nt << 4) + (lane & 15);
      v8f acc;
      const float bias = nt ? b2v1 : b2v0;
      #pragma unroll
      for (int i = 0; i < 8; ++i) acc[i] = bias;
      #pragma unroll
      for (int ks = 0; ks < 8; ++ks) {
        const int kk = (ks << 2) + koff;
        v2f a = *(const v2f*)&s_h[m * HID_ + kk];
        v2f w; w.x = s_w2[kk * G_ + gn]; w.y = s_w2[(kk + 1) * G_ + gn];
        acc = wmma_f32(a, w, acc);
      }
      #pragma unroll
      for (int i = 0; i < 8; ++i) z2[(size_t)(mb + i) * G_ + gn] = acc[i];
    }
    // Buffer t&1 is re-filled only in iteration t+1, i.e. after every
    // thread passed barrier (c) of iteration t -> phase-1 reads are done.
  }
}

// ---------------------------------------------------------------------------
// K3: LSTM2 over full sequence, keep only last h; write h ++ inputs2 to out.
// ---------------------------------------------------------------------------
__global__ __launch_bounds__(128) void k_lstm2(
    const float* __restrict__ zx2, const float* __restrict__ U2,
    const float* __restrict__ in2, float* __restrict__ out)
{
  __shared__ float s_zx[2][16 * G_];
  __shared__ float s_h[16 * HID_];
  __shared__ float s_c[16 * HID_];
  __shared__ float s_z[16 * G_];
  __shared__ float s_u[HID_ * G_];

  const int b0   = blockIdx.x * 16;
  const int tid  = threadIdx.x;
  const int lane = tid & 31;
  const int wave = tid >> 5;
  const int m    = lane & 15;
  const int koff = (lane >> 4) << 1;
  const int mb   = (lane >> 4) << 3;

  for (int i = tid; i < HID_ * G_; i += 128) s_u[i] = U2[i];
  for (int i = tid; i < 16 * HID_; i += 128) { s_h[i] = 0.0f; s_c[i] = 0.0f; }

  issue_tile(zx2 + (size_t)b0 * G_, s_zx[0], tid);
  __syncthreads();

  for (int t = 0; t < TC; ++t) {
    const float* zl = s_zx[t & 1];
    if (t + 1 < TC) {
      issue_tile(zx2 + ((size_t)(t + 1) * B_ + b0) * G_, s_zx[(t + 1) & 1], tid);
      wait_async4();
    } else {
      wait_async0();
    }
    __syncthreads();

    #pragma unroll
    for (int nt = 0; nt < 2; ++nt) {
      const int gn = (wave << 5) + (nt << 4) + (lane & 15);
      v8f acc;
      #pragma unroll
      for (int i = 0; i < 8; ++i) acc[i] = zl[(mb + i) * G_ + gn];
      #pragma unroll
      for (int ks = 0; ks < 8; ++ks) {
        const int kk = (ks << 2) + koff;
        v2f a = *(const v2f*)&s_h[m * HID_ + kk];
        v2f w; w.x = s_u[kk * G_ + gn]; w.y = s_u[(kk + 1) * G_ + gn];
        acc = wmma_f32(a, w, acc);
      }
      #pragma unroll
      for (int i = 0; i < 8; ++i) s_z[(mb + i) * G_ + gn] = acc[i];
    }
    __syncthreads();

    #pragma unroll
    for (int k = 0; k < 4; ++k) {
      const int idx = tid + (k << 7);
      const int mm = idx >> 5, j = idx & 31;
      const float zi = s_z[mm * G_ + j];
      const float zf = s_z[mm * G_ + HID_ + j];
      const float zg = s_z[mm * G_ + 2 * HID_ + j];
      const float zo = s_z[mm * G_ + 3 * HID_ + j];
      const float c = sigm(zf) * s_c[idx] + sigm(zi) * tanhf(zg);
      s_c[idx] = c;
      s_h[idx] = sigm(zo) * tanhf(c);
    }
    __syncthreads();
  }

  // out[b, 0:32] = h_T ; out[b, 32:48] = inputs2[b]
  for (int i = tid; i < 16 * HID_; i += 128) {
    const int mm = i >> 5, j = i & 31;
    out[(size_t)(b0 + mm) * 48 + j] = s_h[i];
  }
  for (int i = tid; i < 16 * 16; i += 128) {
    const int mm = i >> 4, j = i & 15;
    out[(size_t)(b0 + mm) * 48 + 32 + j] = in2[(size_t)(b0 + mm) * 16 + j];
  }
}

// ---------------------------------------------------------------------------
extern "C" void kernel_launch(void* const* d_in, const int* in_sizes, int n_in,
                              void* d_out, int out_size, void* d_ws, size_t ws_size,
                              hipStream_t stream) {
  const float* x1 = (const float*)d_in[0];   // (512, 1024, 32)
  const float* x2 = (const float*)d_in[1];   // (512, 16)
  const float* cw = (const float*)d_in[2];   // (3, 32, 64)
  const float* cb = (const float*)d_in[3];   // (64,)
  const float* W1 = (const float*)d_in[4];   // (64, 128)
  const float* U1 = (const float*)d_in[5];   // (32, 128)
  const float* b1 = (const float*)d_in[6];   // (128,)
  const float* W2 = (const float*)d_in[7];   // (32, 128)
  const float* U2 = (const float*)d_in[8];   // (32, 128)
  const float* b2 = (const float*)d_in[9];   // (128,)
  float* out = (float*)d_out;

  const size_t zxElems = (size_t)TC * B_ * G_;   // 66,977,792 floats each
  float* zx1 = (float*)d_ws;
  float* zx2 = zx1 + zxElems;

  dim3 g1(64, B_);   // 64 time tiles x 512 batch rows
  k_conv_zx1<<<g1, 128, 0, stream>>>(x1, cw, cb, W1, b1, zx1);
  k_lstm1<<<B_ / 16, 128, 0, stream>>>(zx1, U1, W2, b2, zx2);
  k_lstm2<<<B_ / 16, 128, 0, stream>>>(zx2, U2, x2, out);
}